// ChamferBoundarySDFLoss_66864050864913
// MI455X (gfx1250) — compile-verified
//
#include <hip/hip_runtime.h>
#include <hip/hip_bf16.h>
#include <math.h>

typedef __attribute__((ext_vector_type(2))) float v2f;
typedef __attribute__((ext_vector_type(8))) float v8f;

#define BATCH 2
#define HGT 80
#define WID 80
#define HW (HGT*WID)
#define NV ((HGT-1)*WID)          // 6320 vertical edge points
#define NPTS (NV + HGT*(WID-1))   // 12640 total points per grid
#define NTILE (NPTS/16)           // 790 tiles of 16 (exact)
#define EPSV 1e-8f
// Sentinel for invalid points: distance valid<->sentinel ~= sqrt(2)*SENT = 1e10 = BIG
#define SENT 7.0710678e9f

// ---------------------------------------------------------------------------
// Kernel 1: pixel L1 loss -> d_out[0].  Single block, LDS tree reduction.
// ---------------------------------------------------------------------------
__global__ __launch_bounds__(256)
void pixel_loss_kernel(const float* __restrict__ pred,
                       const float* __restrict__ gt,
                       float* __restrict__ d_out) {
    __shared__ float ss[256];
    int t = threadIdx.x;
    float s = 0.0f;
    for (int i = t; i < BATCH*HW; i += 256) s += fabsf(pred[i] - gt[i]);
    ss[t] = s; __syncthreads();
    for (int off = 128; off; off >>= 1) {
        if (t < off) ss[t] += ss[t + off];
        __syncthreads();
    }
    if (t == 0) d_out[0] = ss[0] / (float)(BATCH*HW);   // PIXEL_W = 1.0
}

// ---------------------------------------------------------------------------
// Kernel 2: zero-crossing extraction for 4 grids (2 batches x {pred,gt}).
// Writes WMMA-ready prep vectors:
//   prepA[g][m] = [x, y, x^2+y^2, 1]        (A-side row, K=4)
//   prepB[g][n] = [-2x, -2y, 1, x^2+y^2]    (B-side col, K=4)
// so that A-row . B-col = |p|^2 + |g|^2 - 2 p.g = d^2  (matches reference).
// Invalid points are moved to the sentinel (SENT,SENT): any valid<->invalid
// pair gets d ~= 1e10 (reference's BIG), so no masking is needed inside the
// WMMA loop; masked rows are simply excluded from the finalize sums.
// ---------------------------------------------------------------------------
__global__ __launch_bounds__(256)
void extract_kernel(const float* __restrict__ pred,
                    const float* __restrict__ gt,
                    float* __restrict__ prepA,
                    float* __restrict__ prepB,
                    float* __restrict__ msk) {
    int gid = blockIdx.x * blockDim.x + threadIdx.x;
    if (gid >= 4*NPTS) return;
    int g = gid / NPTS, e = gid - g*NPTS;
    int b = g >> 1, which = g & 1;
    const float* s = (which ? gt : pred) + b*HW;

    float x, y, m;
    if (e < NV) {                       // vertical pairs: (i,j)-(i+1,j)
        int i = e / WID, j = e - i*WID;
        float v1 = s[i*WID + j], v2 = s[(i+1)*WID + j];
        bool c1 = (v1 == 0.0f);
        bool c2 = !c1 && (v2 == 0.0f);
        bool cx = !c1 && !c2 && (v1*v2 < 0.0f);
        float alpha = fabsf(v1) / (fabsf(v1) + fabsf(v2) + EPSV);
        x = c1 ? (float)i : (c2 ? (float)(i+1) : (float)i + alpha);
        y = (float)j;
        m = (c1 || c2 || cx) ? 1.0f : 0.0f;
    } else {                            // horizontal pairs: (i,j)-(i,j+1)
        int e2 = e - NV;
        int i = e2 / (WID-1), j = e2 - i*(WID-1);
        float h1 = s[i*WID + j], h2 = s[i*WID + j + 1];
        bool d1 = (h1 == 0.0f);
        bool d2 = !d1 && (h2 == 0.0f);
        bool dx = !d1 && !d2 && (h1*h2 < 0.0f);
        float beta = fabsf(h1) / (fabsf(h1) + fabsf(h2) + EPSV);
        x = (float)i;
        y = d1 ? (float)j : (d2 ? (float)(j+1) : (float)j + beta);
        m = (d1 || d2 || dx) ? 1.0f : 0.0f;
    }
    if (m == 0.0f) { x = SENT; y = SENT; }     // park invalid points far away
    float n2 = x*x + y*y;
    float* pa = prepA + ((size_t)g*NPTS + e)*4;
    pa[0] = x;        pa[1] = y;        pa[2] = n2;   pa[3] = 1.0f;
    float* pb = prepB + ((size_t)g*NPTS + e)*4;
    pb[0] = -2.0f*x;  pb[1] = -2.0f*y;  pb[2] = 1.0f; pb[3] = n2;
    msk[(size_t)g*NPTS + e] = m;
}

// ---------------------------------------------------------------------------
// Kernel 3: row-min of pairwise d^2 via V_WMMA_F32_16X16X4_F32.
// One wave owns one 16-row tile (m0..m0+15) and streams all 790 column tiles.
// A-layout (32-bit 16x4): lane l, VGPR v  <-> A[l&15][2*(l>=16)+v]  -> float2
// B-layout mirrors A:     lane l, VGPR v  <-> B[2*(l>=16)+v][l&15]  -> float2
// D-layout (32-bit 16x16): lane l, VGPR v <-> D[v + 8*(l>=16)][l&15]
// Steady-state loop: 1 b64 load + 1 wmma + 8 fmin (min3-fused at unroll-2).
// Clamp d^2>=0 and sqrt are monotone => deferred to the finalize kernel.
// Tile guard is wave-uniform so EXEC is all-ones for every WMMA (ISA req).
// ---------------------------------------------------------------------------
__global__ __launch_bounds__(256)
void chamfer_rowmin_kernel(const float* __restrict__ prepA,   // rows
                           const float* __restrict__ prepB,   // columns
                           float* __restrict__ outMin) {      // NPTS floats: min d^2 per row
    int wave = threadIdx.x >> 5;
    int tile = blockIdx.x * (blockDim.x >> 5) + wave;
    if (tile >= NTILE) return;                 // wave-uniform exit

    int lane = threadIdx.x & 31;
    int half = lane >> 4;                      // 0: lanes 0-15, 1: lanes 16-31
    int l15  = lane & 15;
    int m0   = tile * 16;

    // A tile: per-lane float2 = prepA[(m0+l15)*4 + 2*half .. +1]
    v2f a = *(const v2f*)(prepA + ((size_t)(m0 + l15))*4 + 2*half);

    float acc[8];
#pragma unroll
    for (int v = 0; v < 8; ++v) acc[v] = 3.0e38f;

    const float* bptr = prepB + (size_t)l15*4 + 2*half;
    for (int nt = 0; nt < NTILE; ++nt) {
        v2f bb = *(const v2f*)(bptr + (size_t)nt*64);   // 16 points * 4 floats

        v8f c = {0.f,0.f,0.f,0.f,0.f,0.f,0.f,0.f};
        v8f d = __builtin_amdgcn_wmma_f32_16x16x4_f32(
                    /*neg_a=*/false, a, /*neg_b=*/false, bb,
                    /*c_mod=*/(short)0, c, /*reuse_a=*/false, /*reuse_b=*/false);

#pragma unroll
        for (int v = 0; v < 8; ++v) acc[v] = fminf(acc[v], d[v]);
    }

    // Min across the 16 lanes of each half (columns 0..15 of the tile).
#pragma unroll
    for (int v = 0; v < 8; ++v) {
        float x = acc[v];
        for (int mk = 8; mk; mk >>= 1) x = fminf(x, __shfl_xor(x, mk, 32));
        acc[v] = x;
    }
    if (l15 == 0) {
#pragma unroll
        for (int v = 0; v < 8; ++v) outMin[m0 + v + 8*half] = acc[v];
    }
}

// ---------------------------------------------------------------------------
// Kernel 4: finalize.  mins layout: array a = b*2+side (side 0: pred-point
// mins, side 1: gt-point mins) — identical indexing to the mask arrays.
// chamfer_b = -sum(sqrt(minP))/max(np,1) + sum(sqrt(minG))/max(ng,1)
// d_out[1]  = mean over batches.  Clamp d^2>=0 applied here (monotone).
// ---------------------------------------------------------------------------
__global__ __launch_bounds__(256)
void finalize_kernel(const float* __restrict__ mins,
                     const float* __restrict__ msk,
                     float* __restrict__ d_out) {
    __shared__ float ssum[256];
    __shared__ float scnt[256];
    __shared__ float avg[4];
    int t = threadIdx.x;
    for (int a = 0; a < 4; ++a) {
        float s = 0.0f, c = 0.0f;
        for (int i = t; i < NPTS; i += 256) {
            float mk = msk[(size_t)a*NPTS + i];
            if (mk != 0.0f) {
                s += sqrtf(fmaxf(mins[(size_t)a*NPTS + i], 0.0f));
                c += 1.0f;
            }
        }
        ssum[t] = s; scnt[t] = c; __syncthreads();
        for (int off = 128; off; off >>= 1) {
            if (t < off) { ssum[t] += ssum[t+off]; scnt[t] += scnt[t+off]; }
            __syncthreads();
        }
        if (t == 0) avg[a] = ssum[0] / fmaxf(scnt[0], 1.0f);
        __syncthreads();
    }
    if (t == 0) {
        float ch0 = -avg[0] + avg[1];
        float ch1 = -avg[2] + avg[3];
        d_out[1] = 0.5f * (ch0 + ch1);
    }
}

// ---------------------------------------------------------------------------
extern "C" void kernel_launch(void* const* d_in, const int* in_sizes, int n_in,
                              void* d_out, int out_size, void* d_ws, size_t ws_size,
                              hipStream_t stream) {
    (void)in_sizes; (void)n_in; (void)out_size; (void)ws_size;
    const float* pred = (const float*)d_in[0];   // (B,1,H,W) f32
    const float* gt   = (const float*)d_in[1];   // (B,1,H,W) f32
    float* out = (float*)d_out;                  // [loss, last_chamfer]

    // Workspace layout (floats): prepA | prepB | mask | mins  (~2.0 MB total)
    float* prepA = (float*)d_ws;
    float* prepB = prepA + (size_t)4*NPTS*4;
    float* msk   = prepB + (size_t)4*NPTS*4;
    float* mins  = msk   + (size_t)4*NPTS;

    pixel_loss_kernel<<<1, 256, 0, stream>>>(pred, gt, out);

    extract_kernel<<<(4*NPTS + 255)/256, 256, 0, stream>>>(pred, gt, prepA, prepB, msk);

    int blocks = (NTILE + 7) / 8;   // 8 waves (row tiles) per 256-thread block
    for (int b = 0; b < BATCH; ++b) {
        int gp = b*2 + 0;   // pred grid
        int gg = b*2 + 1;   // gt grid
        // min over gt columns for each pred point -> mins[gp]
        chamfer_rowmin_kernel<<<blocks, 256, 0, stream>>>(
            prepA + (size_t)gp*NPTS*4, prepB + (size_t)gg*NPTS*4,
            mins + (size_t)gp*NPTS);
        // min over pred columns for each gt point -> mins[gg]
        chamfer_rowmin_kernel<<<blocks, 256, 0, stream>>>(
            prepA + (size_t)gg*NPTS*4, prepB + (size_t)gp*NPTS*4,
            mins + (size_t)gg*NPTS);
    }

    finalize_kernel<<<1, 256, 0, stream>>>(mins, msk, out);
}